// SparseHealpixSampler_20486994002609
// MI455X (gfx1250) — compile-verified
//
#include <hip/hip_runtime.h>
#include <stdint.h>

// Problem constants (from reference setup_inputs)
#define B_ 2
#define N_ 8192
#define M_ 12288
#define F_ 4
#define H_ 32
#define K_ 64

typedef __attribute__((ext_vector_type(2))) float v2f;
typedef __attribute__((ext_vector_type(8))) float v8f;

// ---------- 64-bit wave shuffles (wave32) ----------
__device__ __forceinline__ unsigned long long shfl64(unsigned long long v, int src) {
    unsigned lo = (unsigned)__shfl((int)(v & 0xffffffffu), src, 32);
    unsigned hi = (unsigned)__shfl((int)(v >> 32), src, 32);
    return ((unsigned long long)hi << 32) | lo;
}
__device__ __forceinline__ unsigned long long shflxor64(unsigned long long v, int m) {
    unsigned lo = (unsigned)__shfl_xor((int)(v & 0xffffffffu), m, 32);
    unsigned hi = (unsigned)__shfl_xor((int)(v >> 32), m, 32);
    return ((unsigned long long)hi << 32) | lo;
}

// ---------- order-preserving key: (value descending, index ascending) ----------
__device__ __forceinline__ unsigned long long make_key(float val, int idx) {
    unsigned u = __float_as_uint(val);
    u ^= (u >> 31) ? 0xFFFFFFFFu : 0x80000000u;   // monotone float->uint (ascending)
    unsigned d = ~u;                               // descending value
    return ((unsigned long long)d << 32) | (unsigned)idx;
}
__device__ __forceinline__ float key_val(unsigned long long key) {
    unsigned u = ~(unsigned)(key >> 32);
    unsigned bits = (u & 0x80000000u) ? (u ^ 0x80000000u) : (u ^ 0xFFFFFFFFu);
    return __uint_as_float(bits);
}

// ---------- kernel 1: spherical -> cartesian, padded to float4 ----------
__global__ __launch_bounds__(256) void sph2cart_kernel(
    const float* __restrict__ los_tp, const float* __restrict__ pix_tp,
    float4* __restrict__ rlos, float4* __restrict__ rpix)
{
    int i = blockIdx.x * 256 + threadIdx.x;
    if (i < B_ * N_) {
        float th = los_tp[2 * i], ph = los_tp[2 * i + 1];
        float st = sinf(th);
        rlos[i] = make_float4(st * cosf(ph), st * sinf(ph), cosf(th), 0.0f);
    } else if (i < B_ * N_ + M_) {
        int j = i - B_ * N_;
        float th = pix_tp[2 * j], ph = pix_tp[2 * j + 1];
        float st = sinf(th);
        rpix[j] = make_float4(st * cosf(ph), st * sinf(ph), cosf(th), 0.0f);
    }
}

// ---------- kernel 2: one wave per pixel: stream N, top-64, WMMA MLP, softmax pool ----------
__global__ __launch_bounds__(256) void knn_pool_kernel(
    const float* __restrict__ x,            // [B,N,4]
    const float4* __restrict__ rlos,        // [B,N] float4
    const float4* __restrict__ rpix,        // [M]   float4
    const float* __restrict__ W1,           // [5,32] row-major
    const float* __restrict__ b1,           // [32]
    const float* __restrict__ W2,           // [32]
    const float* __restrict__ b2,           // [1] (softmax-invariant; unused)
    float* __restrict__ pooled_out,         // [B,M,4]
    int* __restrict__ idx_out)              // [B,M,64]
{
    __shared__ uint2 s_ent[8 * K_];   // per-wave sorted (idx, valbits)
    __shared__ float s_lgt[8 * K_];   // per-wave logits by rank
    (void)b2;

    const int lane = threadIdx.x & 31;
    const int wid  = threadIdx.x >> 5;
    const int p = blockIdx.x * 8 + wid;        // pixel in [0, B*M)
    const int b = p / M_;
    const int m = p - b * M_;

    // ---- phase 1: streaming top-64 over N, register-resident ----
    const float4 rp = rpix[m];
    const float4* losb = rlos + (size_t)b * N_;
    unsigned long long kept0 = ~0ull, kept1 = ~0ull;   // slots v=lane, v=lane+32
    unsigned long long worstKey = ~0ull;
    int worstSlot = 63;

    for (int base = 0; base < N_; base += 32) {
        int n = base + lane;
        float4 rl = losb[n];
        float c = fmaf(rp.x, rl.x, fmaf(rp.y, rl.y, rp.z * rl.z));
        c = fminf(1.0f, fmaxf(-1.0f, c));
        unsigned long long key = make_key(c, n);
        unsigned long long mask = __ballot(key < worstKey);
        while (mask) {                                   // rare after warm-up
            int s = __ffsll(mask) - 1;
            mask &= mask - 1;
            unsigned long long ck = shfl64(key, s);
            if (ck < worstKey) {                         // uniform condition
                if (lane == (worstSlot & 31)) {
                    if (worstSlot < 32) kept0 = ck; else kept1 = ck;
                }
                // wave arg-max over kept -> new worst (uniform result)
                unsigned long long bk; int bs;
                if (kept0 >= kept1) { bk = kept0; bs = lane; }
                else                { bk = kept1; bs = lane + 32; }
                for (int off = 16; off > 0; off >>= 1) {
                    unsigned long long ok = shflxor64(bk, off);
                    int os = __shfl_xor(bs, off, 32);
                    if (ok > bk || (ok == bk && os > bs)) { bk = ok; bs = os; }
                }
                worstKey = bk; worstSlot = bs;
            }
        }
    }

    // ---- phase 2: bitonic sort 64 keys ascending (== value desc, idx asc) ----
    for (int k = 2; k <= 64; k <<= 1) {
        for (int j = k >> 1; j > 0; j >>= 1) {
            if (j == 32) {                               // only k==64: all ascending
                unsigned long long lo = kept0 < kept1 ? kept0 : kept1;
                unsigned long long hi = kept0 < kept1 ? kept1 : kept0;
                kept0 = lo; kept1 = hi;
            } else {
                {
                    unsigned long long o = shflxor64(kept0, j);
                    bool takeMin = (((lane & k) == 0) == ((lane & j) == 0));
                    kept0 = takeMin ? (kept0 < o ? kept0 : o) : (kept0 > o ? kept0 : o);
                }
                {
                    int v = lane + 32;
                    unsigned long long o = shflxor64(kept1, j);
                    bool takeMin = (((v & k) == 0) == ((v & j) == 0));
                    kept1 = takeMin ? (kept1 < o ? kept1 : o) : (kept1 > o ? kept1 : o);
                }
            }
        }
    }

    const int idx0 = (int)(kept0 & 0xffffffffu);
    const int idx1 = (int)(kept1 & 0xffffffffu);
    const float val0 = key_val(kept0);
    const float val1 = key_val(kept1);

    const int rowbase = p * K_;
    idx_out[rowbase + lane]      = idx0;
    idx_out[rowbase + 32 + lane] = idx1;

    const int wb = wid * K_;
    s_ent[wb + lane]      = make_uint2((unsigned)idx0, __float_as_uint(val0));
    s_ent[wb + 32 + lane] = make_uint2((unsigned)idx1, __float_as_uint(val1));
    __syncthreads();

    // ---- phase 3: MLP via V_WMMA_F32_16X16X4_F32 ----
    // h[16t+m, j] = relu( sum_k x[idx,k]*W1[k,j] + d*W1[4,j] + b1[j] )
    const int half = lane >> 4;
    const int l    = lane & 15;
    const float4* x4 = (const float4*)x;
    const size_t xbase = (size_t)b * N_;

    v2f bt[2];
    float b1v[2], w4v[2], w2v[2];
#pragma unroll
    for (int ct = 0; ct < 2; ++ct) {
        int j = l + 16 * ct;
        int k0 = half ? 2 : 0;                 // B layout: lane half picks K pair
        bt[ct].x = W1[k0 * H_ + j];
        bt[ct].y = W1[(k0 + 1) * H_ + j];
        b1v[ct] = b1[j];
        w4v[ct] = W1[4 * H_ + j];
        w2v[ct] = W2[j];
    }

#pragma unroll
    for (int t = 0; t < 4; ++t) {
        // A: 16x4 tile of x features for ranks 16t..16t+15
        uint2 e = s_ent[wb + 16 * t + l];
        float4 xf = x4[xbase + e.x];
        v2f a;
        if (half == 0) { a.x = xf.x; a.y = xf.y; }     // K=0,1
        else           { a.x = xf.z; a.y = xf.w; }     // K=2,3

        // d (= arccos(val)) for the 8 rows this lane accumulates
        float dm[8];
#pragma unroll
        for (int r = 0; r < 8; ++r) {
            uint2 er = s_ent[wb + 16 * t + r + 8 * half];
            float v = __uint_as_float(er.y);
            dm[r] = acosf(fminf(1.0f, fmaxf(-1.0f, v)));
        }

        float lp[8];
#pragma unroll
        for (int r = 0; r < 8; ++r) lp[r] = 0.0f;

#pragma unroll
        for (int ct = 0; ct < 2; ++ct) {
            v8f c;
#pragma unroll
            for (int r = 0; r < 8; ++r) c[r] = fmaf(dm[r], w4v[ct], b1v[ct]);
            v8f D = __builtin_amdgcn_wmma_f32_16x16x4_f32(
                false, a, false, bt[ct], (short)0, c, false, false);
#pragma unroll
            for (int r = 0; r < 8; ++r)
                lp[r] = fmaf(fmaxf(D[r], 0.0f), w2v[ct], lp[r]);   // relu + @W2
        }
        // reduce logits across the 16 lanes of each half (columns j)
#pragma unroll
        for (int r = 0; r < 8; ++r) {
            for (int off = 8; off > 0; off >>= 1)
                lp[r] += __shfl_xor(lp[r], off, 32);
        }
        if (l == 0) {
#pragma unroll
            for (int r = 0; r < 8; ++r)
                s_lgt[wb + 16 * t + r + 8 * half] = lp[r];
        }
    }
    __syncthreads();

    // ---- phase 4: softmax over K=64 + weighted pooling ----
    float l0 = s_lgt[wb + lane];
    float l1 = s_lgt[wb + 32 + lane];
    float mx = fmaxf(l0, l1);
    for (int off = 16; off > 0; off >>= 1)
        mx = fmaxf(mx, __shfl_xor(mx, off, 32));
    float e0 = expf(l0 - mx), e1 = expf(l1 - mx);
    float sum = e0 + e1;
    for (int off = 16; off > 0; off >>= 1)
        sum += __shfl_xor(sum, off, 32);
    float w0 = e0 / sum, w1 = e1 / sum;

    float4 xa = x4[xbase + (unsigned)idx0];
    float4 xbv = x4[xbase + (unsigned)idx1];
    float a0 = fmaf(w0, xa.x, w1 * xbv.x);
    float a1 = fmaf(w0, xa.y, w1 * xbv.y);
    float a2 = fmaf(w0, xa.z, w1 * xbv.z);
    float a3 = fmaf(w0, xa.w, w1 * xbv.w);
    for (int off = 16; off > 0; off >>= 1) {
        a0 += __shfl_xor(a0, off, 32);
        a1 += __shfl_xor(a1, off, 32);
        a2 += __shfl_xor(a2, off, 32);
        a3 += __shfl_xor(a3, off, 32);
    }
    if (lane == 0)
        ((float4*)pooled_out)[p] = make_float4(a0, a1, a2, a3);
}

extern "C" void kernel_launch(void* const* d_in, const int* in_sizes, int n_in,
                              void* d_out, int out_size, void* d_ws, size_t ws_size,
                              hipStream_t stream) {
    (void)in_sizes; (void)n_in; (void)out_size; (void)ws_size;
    const float* x   = (const float*)d_in[0];
    const float* los = (const float*)d_in[1];
    const float* pix = (const float*)d_in[2];
    const float* W1  = (const float*)d_in[3];
    const float* b1  = (const float*)d_in[4];
    const float* W2  = (const float*)d_in[5];
    const float* b2  = (const float*)d_in[6];

    float4* rlos = (float4*)d_ws;                      // [B*N] float4 (256 KB)
    float4* rpix = rlos + (size_t)B_ * N_;             // [M]   float4 (192 KB)

    float* pooled = (float*)d_out;                                   // [B,M,4] f32
    int*   idxout = (int*)((float*)d_out + (size_t)B_ * M_ * F_);    // [B,M,64] i32

    int totalPrep = B_ * N_ + M_;
    sph2cart_kernel<<<(totalPrep + 255) / 256, 256, 0, stream>>>(los, pix, rlos, rpix);

    knn_pool_kernel<<<(B_ * M_) / 8, 256, 0, stream>>>(
        x, rlos, rpix, W1, b1, W2, b2, pooled, idxout);
}